// compute_feature_16045997818334
// MI455X (gfx1250) — compile-verified
//
#include <hip/hip_runtime.h>

// CDNA5 / gfx1250: chunked-scan-as-matmul IIR cascade, segment-parallel.
//
//   y_t   = 0.939*y_{t-1} + (w_t - w_{t-1} + eps)
//   char  = y^2 + 3*diff(y)^2
//   s_t   = 0.4*s_{t-1}   + 0.6*char_t
//   l_t   = 0.985*l_{t-1} + 0.015*char_t
//
// Each 16-step chunk of each recurrence is a lower-triangular decay matmul
// (4 accumulating v_wmma_f32_16x16x4_f32, K=16). First-diff + eps are folded
// into the raw-IIR matrix + C-operand boundary terms, so only the raw wave
// tile is loaded. Time axis is split into 15 segments with a 1280-step
// zero-init warm-up (slowest filter decays 0.985^1280 ~ 4e-9 < f32 eps),
// giving 720 independent waves. 3 waves per block (one per channel) stage
// the [16b][16t][12] output tile in LDS and drain it with coalesced b128
// stores.

typedef __attribute__((ext_vector_type(2))) float v2f;
typedef __attribute__((ext_vector_type(8))) float v8f;

#define RAW_C  0.939f
#define EPSV   1e-10f
#define STA_A  0.4f      // 1 - 0.6
#define STA_W  0.6f
#define LTA_A  0.985f    // 1 - 0.015
#define LTA_W  0.015f
#define TLEN   30000
#define NBATCH 256
#define SEGS   15
#define SEGLEN 2000      // TLEN / SEGS, multiple of 16
#define WARM   1280      // warm-up steps (80 chunks), 0.985^1280 ~ 4e-9
#define LDSROW 196       // 16*12 + 4 pad floats (bank-conflict break)

__device__ __forceinline__ float powi(float a, int e) {
    float r = 1.0f;
#pragma unroll 1
    for (int i = 0; i < e; ++i) r *= a;
    return r;
}

__global__ __launch_bounds__(96)
void iir_wmma_kernel(const float* __restrict__ wave, float* __restrict__ out) {
    const int tid  = threadIdx.x;
    const int lane = tid & 31;             // wave32 lane
    const int ch   = tid >> 5;             // one channel per wave
    const int lid  = lane & 15;            // column (batch) / M index
    const int hi   = lane >> 4;            // lane-half

    const int bt = blockIdx.x / SEGS;      // batch tile 0..15
    const int sg = blockIdx.x % SEGS;      // time segment 0..14
    const int b0 = bt * 16;
    const int b  = b0 + lid;

    const int outStart = sg * SEGLEN;
    const int tBegin   = (sg == 0) ? 0 : (outStart - WARM);
    const int tEnd     = outStart + SEGLEN;

    const float* colBase = wave + ((size_t)b * TLEN) * 3 + ch;  // stride 3 along t

    __shared__ float lds[16 * LDSROW];     // [16 batches][196] staging tile

    // ---- constant 16x16 matrices as A operands (16x4 blocks) ----
    // A block q, comp p: lane L holds (M = L&15, K = 4q + p + 2*(L>>4)).
    // A1 = raw-IIR matrix with first-diff folded in:
    //   A1[t,t] = 1 ; A1[t,j<t] = c^(t-j-1)*(c-1)
    v2f A1[4], A2[4], A3[4];
#pragma unroll
    for (int q = 0; q < 4; ++q) {
#pragma unroll
        for (int p = 0; p < 2; ++p) {
            int K = 4 * q + p + 2 * hi;
            A1[q][p] = (lid == K) ? 1.0f
                     : (lid >  K) ? powi(RAW_C, lid - K - 1) * (RAW_C - 1.0f)
                                  : 0.0f;
            A2[q][p] = (lid >= K) ? STA_W * powi(STA_A, lid - K) : 0.0f;
            A3[q][p] = (lid >= K) ? LTA_W * powi(LTA_A, lid - K) : 0.0f;
        }
    }
    // C-operand boundary factors, row M = r + 8*hi:
    //  f1 = c^(M+1) (y carry), fc = c^M (w_last), fe = eps*(1-c^(M+1))/(1-c)
    float f1[8], fc[8], fe[8], f2[8], f3[8];
#pragma unroll
    for (int r = 0; r < 8; ++r) {
        int M = r + 8 * hi;
        f1[r] = powi(RAW_C, M + 1);
        fc[r] = powi(RAW_C, M);
        fe[r] = EPSV * (1.0f - f1[r]) / (1.0f - RAW_C);
        f2[r] = powi(STA_A, M + 1);
        f3[r] = powi(LTA_A, M + 1);
    }

    float yprev = 0.0f, sprev = 0.0f, lprev = 0.0f;
    float wlast = (tBegin > 0) ? colBase[(size_t)(tBegin - 1) * 3] : 0.0f;

    for (int t0 = tBegin; t0 < tEnd; t0 += 16) {
        const bool emit = (t0 >= outStart);          // block-uniform
        if (t0 + 16 < tEnd)
            __builtin_prefetch(colBase + (size_t)(t0 + 16) * 3, 0, 1);

        // ---- load raw wave tile directly in B layout ----
        // B block q, comp p: lane L holds (K = 4q + p + 2*(L>>4), N = L&15)
        v2f WB[4];
#pragma unroll
        for (int q = 0; q < 4; ++q) {
#pragma unroll
            for (int p = 0; p < 2; ++p) {
                int t = t0 + 4 * q + p + 2 * hi;
                WB[q][p] = colBase[(size_t)t * 3];
            }
        }

        // ---- Y = (c^{M+1} yprev - c^M wlast + eps row) + M1 * W ----
        v8f Y;
#pragma unroll
        for (int r = 0; r < 8; ++r)
            Y[r] = f1[r] * yprev - fc[r] * wlast + fe[r];
#pragma unroll
        for (int q = 0; q < 4; ++q)
            Y = __builtin_amdgcn_wmma_f32_16x16x4_f32(false, A1[q], false, WB[q],
                                                      (short)0, Y, false, false);

        // ---- char = Y^2 + 3*diff(Y)^2  (D layout: VGPR r -> row r + 8*hi) ----
        float y7lo = __shfl(Y[7], lid, 32);          // row 7 -> hi half (M=8 prev)
        float CH[8];
#pragma unroll
        for (int r = 0; r < 8; ++r) {
            float prev = (r == 0) ? (hi ? y7lo : yprev) : Y[r - 1];
            float d    = Y[r] - prev;
            CH[r] = Y[r] * Y[r] + 3.0f * d * d;
        }

        // ---- char: D layout -> B layout (cross-half shuffles) ----
        v2f CB[4];
#pragma unroll
        for (int q = 0; q < 4; ++q) {
#pragma unroll
            for (int p = 0; p < 2; ++p) {
                const int rowLo = 4 * q + p;         // needed by lanes 0-15
                const int rowHi = 4 * q + 2 + p;     // needed by lanes 16-31
                float vlo = __shfl(CH[rowLo & 7], lid + 16 * (rowLo >> 3), 32);
                float vhi = __shfl(CH[rowHi & 7], lid + 16 * (rowHi >> 3), 32);
                CB[q][p] = hi ? vhi : vlo;
            }
        }

        // ---- STA / LTA EMAs over char, same chunked-matmul trick ----
        v8f S, L;
#pragma unroll
        for (int r = 0; r < 8; ++r) { S[r] = f2[r] * sprev; L[r] = f3[r] * lprev; }
#pragma unroll
        for (int q = 0; q < 4; ++q) {
            S = __builtin_amdgcn_wmma_f32_16x16x4_f32(false, A2[q], false, CB[q],
                                                      (short)0, S, false, false);
            L = __builtin_amdgcn_wmma_f32_16x16x4_f32(false, A3[q], false, CB[q],
                                                      (short)0, L, false, false);
        }

        // ---- update carries (row 15 lives in hi-half VGPR 7; w row 15 in WB[3][1]) ----
        yprev = __shfl(Y[7],     lid + 16, 32);
        sprev = __shfl(S[7],     lid + 16, 32);
        lprev = __shfl(L[7],     lid + 16, 32);
        wlast = __shfl(WB[3][1], lid + 16, 32);

        // ---- stage tile in LDS, drain with coalesced b128 stores ----
        if (emit) {
#pragma unroll
            for (int r = 0; r < 8; ++r) {            // char/s/l in D layout
                int base = lid * LDSROW + (r + 8 * hi) * 12;
                lds[base + ch]     = CH[r];
                lds[base + 3 + ch] = S[r];
                lds[base + 6 + ch] = L[r];
            }
#pragma unroll
            for (int q = 0; q < 4; ++q)              // wave passthrough, B layout
#pragma unroll
                for (int p = 0; p < 2; ++p) {
                    int tt = 4 * q + p + 2 * hi;
                    lds[lid * LDSROW + tt * 12 + 9 + ch] = WB[q][p];
                }
            __syncthreads();
            // 16 batches * 192 floats = 768 float4; 96 lanes * 8 each
#pragma unroll
            for (int i = 0; i < 8; ++i) {
                int f  = tid + 96 * i;               // float4 index
                int bb = f / 48;                     // batch row (48 float4 each)
                int r4 = f % 48;
                float4 v = *(const float4*)&lds[bb * LDSROW + r4 * 4];
                float4* gp = reinterpret_cast<float4*>(
                    out + ((size_t)(b0 + bb) * TLEN + t0) * 12 + r4 * 4);
                *gp = v;
            }
            __syncthreads();                         // protect LDS reuse
        }
    }
}

extern "C" void kernel_launch(void* const* d_in, const int* in_sizes, int n_in,
                              void* d_out, int out_size, void* d_ws, size_t ws_size,
                              hipStream_t stream) {
    (void)in_sizes; (void)n_in; (void)out_size; (void)d_ws; (void)ws_size;
    const float* wave = (const float*)d_in[0];
    float*       out  = (float*)d_out;
    // 16 batch tiles x 15 time segments, 3 waves (channels) per block
    dim3 grid(16 * SEGS), block(96);
    hipLaunchKernelGGL(iir_wmma_kernel, grid, block, 0, stream, wave, out);
}